// GCN_dgl_19026705121765
// MI455X (gfx1250) — compile-verified
//
#include <hip/hip_runtime.h>

#define NH 128

typedef __attribute__((ext_vector_type(2))) float f32x2;
typedef __attribute__((ext_vector_type(8))) float f32x8;

// ---------------- utility kernels ----------------

__global__ void zero_scalar_kernel(float* __restrict__ p, long n) {
    long i = (long)blockIdx.x * blockDim.x + threadIdx.x;
    long stride = (long)gridDim.x * blockDim.x;
    for (; i < n; i += stride) p[i] = 0.0f;
}

__global__ void zero_vec4_kernel(float4* __restrict__ p, long n4) {
    long i = (long)blockIdx.x * blockDim.x + threadIdx.x;
    long stride = (long)gridDim.x * blockDim.x;
    float4 z = make_float4(0.f, 0.f, 0.f, 0.f);
    for (; i < n4; i += stride) p[i] = z;
}

__global__ void degree_kernel(const int* __restrict__ src, const int* __restrict__ dst,
                              unsigned* __restrict__ outdeg, unsigned* __restrict__ indeg,
                              int nE) {
    int e = blockIdx.x * blockDim.x + threadIdx.x;
    if (e < nE) {
        atomicAdd(&outdeg[src[e]], 1u);
        atomicAdd(&indeg[dst[e]], 1u);
    }
}

__global__ void norm_kernel(const unsigned* __restrict__ outdeg,
                            const unsigned* __restrict__ indeg,
                            float* __restrict__ nsrc, float* __restrict__ ndst, int n) {
    int i = blockIdx.x * blockDim.x + threadIdx.x;
    if (i < n) {
        nsrc[i] = rsqrtf(fmaxf((float)outdeg[i], 1.0f));
        ndst[i] = rsqrtf(fmaxf((float)indeg[i], 1.0f));
    }
}

// ---------------- WMMA GEMM: Y[rows x 128] = diag(rowscale) * X[rows x K] @ W[K x 128] ----------------
// Block = 256 threads = 8 waves; each wave owns one 16x16 output tile (16 rows x its 16 cols).
// fp32 WMMA (V_WMMA_F32_16X16X4_F32) keeps full reference precision; GEMM cost is
// negligible vs. the L2-bound scatter phase, so no reason to drop to f16/bf16.
// REQUIREMENT: rows % 16 == 0 (true here: 100000 = 6250*16); grid.x == rows/16.
// This lets the whole kernel run with EXEC all-ones (WMMA requirement) and a branch-free
// epilogue: 8 stores at immediate offsets (row stride 512B) off one base address.
template <int K, bool SCALE>
__global__ __launch_bounds__(256) void gemm_rowscale_wmma(
    const float* __restrict__ X, const float* __restrict__ W,
    const float* __restrict__ rowscale, float* __restrict__ Y) {
    const int lane = threadIdx.x & 31;
    const int half = lane >> 4;   // 0: lanes 0-15, 1: lanes 16-31
    const int l16  = lane & 15;
    const int col  = ((threadIdx.x >> 5) << 4) + l16;  // this wave's N + lane's column
    const int row0 = blockIdx.x << 4;

    // A fragment row for this lane (M = l16 for both lane halves)
    const int aRow = row0 + l16;
    const float scale = SCALE ? rowscale[aRow] : 1.0f;
    const float* __restrict__ arow = X + (size_t)aRow * K;

    f32x8 acc = {};
#pragma unroll 4
    for (int k0 = 0; k0 < K; k0 += 4) {
        // ISA 7.12.2 fp32 A 16x4: lanes 0-15 hold K={k0,k0+1}, lanes 16-31 hold K={k0+2,k0+3}
        const int ka = k0 + (half << 1);
        f32x2 a, b;
        a.x = arow[ka] * scale;
        a.y = arow[ka + 1] * scale;
        // B 4x16: row K striped across lanes; lower lanes cover first K pair, upper the second
        b.x = W[(size_t)ka * NH + col];
        b.y = W[(size_t)(ka + 1) * NH + col];
        acc = __builtin_amdgcn_wmma_f32_16x16x4_f32(
            /*neg_a=*/false, a, /*neg_b=*/false, b,
            /*c_mod=*/(short)0, acc, /*reuse_a=*/false, /*reuse_b=*/false);
    }

    // C/D layout: VGPR r -> M = r (lanes 0-15), M = r+8 (lanes 16-31); N = l16-derived col
    float* __restrict__ yp = Y + (size_t)(row0 + (half << 3)) * NH + col;
#pragma unroll
    for (int r = 0; r < 8; ++r) {
        yp[(size_t)r * NH] = acc[r];
    }
}

// ---------------- edge scatter-add: agg[dst] += Y[src], 128 f32 per edge ----------------
// One wave per edge: coalesced 512B gather, dst uniform across the wave -> coalesced atomics.
__global__ __launch_bounds__(256) void scatter_add_kernel(
    const float* __restrict__ Y, const int* __restrict__ src, const int* __restrict__ dst,
    float* __restrict__ agg, int nE) {
    long t = (long)blockIdx.x * blockDim.x + threadIdx.x;
    int e = (int)(t >> 5);
    if (e >= nE) return;
    int j = ((int)t & 31) << 2;  // 4 floats per lane
    const float4 v = *(const float4*)(Y + (size_t)src[e] * NH + j);
    float* a = agg + (size_t)dst[e] * NH + j;
    atomicAdd(a + 0, v.x);
    atomicAdd(a + 1, v.y);
    atomicAdd(a + 2, v.z);
    atomicAdd(a + 3, v.w);
}

// ---------------- finalize: out = relu(agg * norm_dst + bias) [* norm_src for next layer] ----------------
__global__ __launch_bounds__(256) void finalize_kernel(
    const float* __restrict__ agg, const float* __restrict__ ndst,
    const float* __restrict__ nsrc_or_null, const float* __restrict__ bias,
    float* __restrict__ out, int rows) {
    long t = (long)blockIdx.x * blockDim.x + threadIdx.x;
    if (t >= (long)rows * 32) return;
    int i = (int)(t >> 5);
    int j = ((int)t & 31) << 2;
    const float nd = ndst[i];
    const float s  = nsrc_or_null ? nsrc_or_null[i] : 1.0f;
    const float4 v = *(const float4*)(agg + (size_t)i * NH + j);
    const float4 b = *(const float4*)(bias + j);
    float4 o;
    o.x = fmaxf(v.x * nd + b.x, 0.0f) * s;
    o.y = fmaxf(v.y * nd + b.y, 0.0f) * s;
    o.z = fmaxf(v.z * nd + b.z, 0.0f) * s;
    o.w = fmaxf(v.w * nd + b.w, 0.0f) * s;
    *(float4*)(out + (size_t)i * NH + j) = o;
}

// ---------------- driver ----------------

extern "C" void kernel_launch(void* const* d_in, const int* in_sizes, int n_in,
                              void* d_out, int out_size, void* d_ws, size_t ws_size,
                              hipStream_t stream) {
    const float* feat = (const float*)d_in[0];  // [N, 256]
    const float* W1   = (const float*)d_in[1];  // [256, 128]
    const float* b1   = (const float*)d_in[2];  // [128]
    const float* W2   = (const float*)d_in[3];  // [128, 128]
    const float* b2   = (const float*)d_in[4];  // [128]
    const int* esrc   = (const int*)d_in[5];    // [E]
    const int* edst   = (const int*)d_in[6];    // [E]

    const int nN = in_sizes[0] / 256;  // 100000 (multiple of 16)
    const int nE = in_sizes[5];        // 1600000

    // workspace layout (floats): outdeg | indeg | nsrc | ndst | bufA(N*128) | bufB(N*128)
    float* ws = (float*)d_ws;
    unsigned* outdeg = (unsigned*)ws;
    unsigned* indeg  = outdeg + nN;
    float* nsrc = (float*)(indeg + nN);
    float* ndst = nsrc + nN;
    float* bufA = ndst + nN;
    float* bufB = bufA + (size_t)nN * NH;

    const int TB = 256;
    const long nmat  = (long)nN * NH;
    const int gNode  = (nN + TB - 1) / TB;
    const int gEdge  = (nE + TB - 1) / TB;
    const int gZero4 = (int)((nmat / 4 + TB - 1) / TB);
    const int gGemm  = nN / 16;  // rows % 16 == 0
    const int gScat  = (int)(((long)nE * 32 + TB - 1) / TB);
    const int gFin   = (int)(((long)nN * 32 + TB - 1) / TB);

    // degrees + norms
    zero_scalar_kernel<<<gNode, TB, 0, stream>>>((float*)outdeg, 2L * nN);
    degree_kernel<<<gEdge, TB, 0, stream>>>(esrc, edst, outdeg, indeg, nE);
    norm_kernel<<<gNode, TB, 0, stream>>>(outdeg, indeg, nsrc, ndst, nN);

    // ---- layer 1 ----
    // bufA = (feat * nsrc) @ W1
    gemm_rowscale_wmma<256, true><<<gGemm, TB, 0, stream>>>(feat, W1, nsrc, bufA);
    // bufB = segment_sum over edges
    zero_vec4_kernel<<<gZero4, TB, 0, stream>>>((float4*)bufB, nmat / 4);
    scatter_add_kernel<<<gScat, TB, 0, stream>>>(bufA, esrc, edst, bufB, nE);
    // bufA = relu(bufB * ndst + b1) * nsrc   (pre-scaled input of layer 2)
    finalize_kernel<<<gFin, TB, 0, stream>>>(bufB, ndst, nsrc, b1, bufA, nN);

    // ---- layer 2 ----
    // bufB = bufA @ W2
    gemm_rowscale_wmma<128, false><<<gGemm, TB, 0, stream>>>(bufA, W2, nullptr, bufB);
    // bufA = segment_sum over edges
    zero_vec4_kernel<<<gZero4, TB, 0, stream>>>((float4*)bufA, nmat / 4);
    scatter_add_kernel<<<gScat, TB, 0, stream>>>(bufB, esrc, edst, bufA, nE);
    // d_out = relu(bufA * ndst + b2)
    finalize_kernel<<<gFin, TB, 0, stream>>>(bufA, ndst, nullptr, b2, (float*)d_out, nN);
}